// cReLU_percent_40114994544672
// MI455X (gfx1250) — compile-verified
//
#include <hip/hip_runtime.h>
#include <stdint.h>

// Problem constants (from reference): x is (32, 128, 112, 112) f32, k = 64.
#define NCH   128
#define HW    (112 * 112)      // 12544
#define TPB   128              // threads per block (4 waves on wave32)
#define KSEL  64

#ifndef __has_builtin
#define __has_builtin(x) 0
#endif

#if __has_builtin(__builtin_amdgcn_global_load_async_to_lds_b32)
#define HAVE_ASYNC_LDS 1
#else
#define HAVE_ASYNC_LDS 0
#endif

#if __has_builtin(__builtin_amdgcn_s_wait_asynccnt)
#define WAIT_ASYNC() __builtin_amdgcn_s_wait_asynccnt(0)
#else
#define WAIT_ASYNC() asm volatile("s_wait_asynccnt 0" ::: "memory")
#endif

// Builtin signature (probe-confirmed): (global-AS int* src, shared-AS int* dst, imm, imm)
typedef __attribute__((address_space(1))) int* g_i32p;
typedef __attribute__((address_space(3))) int* l_i32p;

__device__ __forceinline__ g_i32p as_global_i32(const float* p) {
    return (g_i32p)(uintptr_t)p;
}
// Flat LDS addresses map to LDS by truncation to the low 32 bits (ISA §10.2).
__device__ __forceinline__ l_i32p as_lds_i32(float* p) {
    return (l_i32p)(uint32_t)(uintptr_t)p;
}

__device__ __forceinline__ void ce_asc(float& lo, float& hi) {
    const float mn = fminf(lo, hi);
    const float mx = fmaxf(lo, hi);
    lo = mn;
    hi = mx;
}

// Batcher odd-even mergesort, ascending, fully unrolled (543 comparators for N=64).
// Standard iterative formulation; all indices compile-time so the array stays in VGPRs.
template <int N>
__device__ __forceinline__ void oddeven_sort_asc(float* a) {
#pragma unroll
    for (int p = 1; p < N; p <<= 1) {
#pragma unroll
        for (int k = p; k >= 1; k >>= 1) {
#pragma unroll
            for (int j = k % p; j + k < N; j += 2 * k) {
#pragma unroll
                for (int i = 0; i < k; ++i) {
                    if (i + j + k < N) {
                        if ((i + j) / (2 * p) == (i + j + k) / (2 * p)) {
                            ce_asc(a[i + j], a[i + j + k]);
                        }
                    }
                }
            }
        }
    }
}

__global__ __launch_bounds__(TPB) void crelu_topk_half_kernel(
    const float* __restrict__ x, float* __restrict__ out) {
    // Per-thread pixel. Blocks never straddle the batch dim: HW % TPB == 0.
    __shared__ float tile[NCH * TPB];  // 64 KB, layout [channel][thread]

    const int t   = threadIdx.x;
    const int p   = blockIdx.x * TPB + t;    // global pixel id
    const int n   = p / HW;
    const int hw  = p - n * HW;
    const size_t base = (size_t)n * NCH * HW + (size_t)hw;

    const float* src = x + base;

    // ---- Phase 1: HBM -> LDS, one touch per element (async data mover) ----
#if HAVE_ASYNC_LDS
#pragma unroll
    for (int c = 0; c < NCH; ++c) {
        __builtin_amdgcn_global_load_async_to_lds_b32(
            as_global_i32(src + (size_t)c * HW),
            as_lds_i32(&tile[c * TPB + t]), 0, 0);
    }
    WAIT_ASYNC();
    asm volatile("" ::: "memory");  // LDS written behind the compiler's back
#else
#pragma unroll
    for (int c = 0; c < NCH; ++c) {
        tile[c * TPB + t] = src[(size_t)c * HW];
    }
#endif

    // ---- Phase 2: exact 64th-largest of 128 via two sorted halves ----
    float a[KSEL];
    float b[KSEL];
#pragma unroll
    for (int c = 0; c < KSEL; ++c) a[c] = tile[c * TPB + t];
#pragma unroll
    for (int c = 0; c < KSEL; ++c) b[c] = tile[(c + KSEL) * TPB + t];

    oddeven_sort_asc<KSEL>(a);
    oddeven_sort_asc<KSEL>(b);

    // Half-cleaner on the bitonic sequence (reverse(A) ++ B):
    // top-64 of (A ∪ B) = { max(a[i], b[63-i]) }; threshold = min of that set.
    float acc[8];
#pragma unroll
    for (int i = 0; i < 8; ++i) acc[i] = fmaxf(a[i], b[KSEL - 1 - i]);
#pragma unroll
    for (int i = 8; i < KSEL; ++i)
        acc[i & 7] = fminf(acc[i & 7], fmaxf(a[i], b[KSEL - 1 - i]));
#pragma unroll
    for (int s = 4; s > 0; s >>= 1) {
#pragma unroll
        for (int i = 0; i < s; ++i) acc[i] = fminf(acc[i], acc[i + s]);
    }
    // Fold the trailing ReLU into the threshold: keep v iff v >= max(thr, 0).
    const float t2 = fmaxf(acc[0], 0.0f);

    // Kill cross-phase CSE so the 128 originals don't stay live through the sort:
    // phase 3 re-reads LDS instead (VGPR peak ~150 instead of ~280).
    asm volatile("" ::: "memory");

    // ---- Phase 3: apply mask from LDS, coalesced non-temporal stores ----
    float* dst = out + base;
#pragma unroll
    for (int c = 0; c < NCH; ++c) {
        const float v = tile[c * TPB + t];
        const float r = (v >= t2) ? v : 0.0f;
        __builtin_nontemporal_store(r, dst + (size_t)c * HW);
    }
}

extern "C" void kernel_launch(void* const* d_in, const int* in_sizes, int n_in,
                              void* d_out, int out_size, void* d_ws, size_t ws_size,
                              hipStream_t stream) {
    const float* x = (const float*)d_in[0];
    float* out = (float*)d_out;

    const int total  = in_sizes[0];        // 32*128*112*112
    const int npix   = total / NCH;        // 401408
    const int blocks = npix / TPB;         // 3136 (exact)

    crelu_topk_half_kernel<<<blocks, TPB, 0, stream>>>(x, out);
}